// sobel_net_1580547967819
// MI455X (gfx1250) — compile-verified
//
#include <hip/hip_runtime.h>
#include <math.h>

// ---------------- problem constants ----------------
constexpr int Bn  = 32;
constexpr int Hn  = 512;
constexpr int Wn  = 512;
constexpr int HWn = Hn * Wn;            // 262144
constexpr size_t CHWn  = 3ull * HWn;    // per-batch input stride
constexpr size_t OCHWn = 4ull * HWn;    // per-batch output stride

// ---------------- tiling ----------------
constexpr int TX = 32;                  // tile width  == wave32 lanes
constexpr int TY = 8;                   // tile height -> 256 threads / 8 waves
constexpr int SW = TX + 2;              // 34 (halo)
constexpr int SH = TY + 2;              // 10 (halo)
constexpr int SN = SW * SH;             // 340 staged gray values
constexpr int NTHREADS = TX * TY;       // 256

// ---------------- CDNA5 async global->LDS path (probe-confirmed: emits
// ---------------- global_load_async_to_lds_b32 + s_wait_asynccnt) ----------
#if defined(__has_builtin)
# if __has_builtin(__builtin_amdgcn_global_load_async_to_lds_b32)
#  define USE_ASYNC_LDS 1
# endif
#endif

typedef __attribute__((address_space(1))) int as1_int;   // "__device__ int"
typedef __attribute__((address_space(3))) int as3_int;   // "__shared__ int"

#define AS1PTR(p) ((as1_int*)(unsigned long long)(p))
#define AS3PTR(p) ((as3_int*)(unsigned int)(unsigned long long)(p))

__device__ __forceinline__ void async_wait_all() {
#if defined(USE_ASYNC_LDS)
# if defined(__has_builtin) && __has_builtin(__builtin_amdgcn_s_wait_asynccnt)
  __builtin_amdgcn_s_wait_asynccnt(0);
# else
  asm volatile("s_wait_asynccnt 0" ::: "memory");
# endif
#endif
}

// Stage (TY+2)x(TX+2) halo tile: R,G,B channels into LDS via async-copy
// (ASYNCcnt path; no VGPR round-trip), then grayscale into LDS. Clamped at
// image borders (clamped values only feed border pixels, whose output is
// forced to zero anyway).
__device__ __forceinline__ void stage_tile(const float* __restrict__ im, int b,
                                           int x0, int y0, int tid,
                                           float* sR, float* sG, float* sB,
                                           float* sGray) {
  const float* base = im + (size_t)b * CHWn;
  for (int i = tid; i < SN; i += NTHREADS) {
    int yy = i / SW;
    int xx = i - yy * SW;
    int gy = y0 - 1 + yy; gy = gy < 0 ? 0 : (gy > Hn - 1 ? Hn - 1 : gy);
    int gx = x0 - 1 + xx; gx = gx < 0 ? 0 : (gx > Wn - 1 ? Wn - 1 : gx);
    size_t p = (size_t)gy * Wn + gx;
#if defined(USE_ASYNC_LDS)
    __builtin_amdgcn_global_load_async_to_lds_b32(AS1PTR(base + p),
                                                  AS3PTR(&sR[i]), 0, 0);
    __builtin_amdgcn_global_load_async_to_lds_b32(AS1PTR(base + HWn + p),
                                                  AS3PTR(&sG[i]), 0, 0);
    __builtin_amdgcn_global_load_async_to_lds_b32(AS1PTR(base + 2 * HWn + p),
                                                  AS3PTR(&sB[i]), 0, 0);
#else
    float r  = base[p];
    float g  = base[HWn + p];
    float bl = base[2 * HWn + p];
    sR[i] = r; sG[i] = g; sB[i] = bl;
    sGray[i] = 0.299f * r + 0.587f * g + 0.114f * bl;
#endif
  }
#if defined(USE_ASYNC_LDS)
  async_wait_all();        // this wave's async writes landed in LDS
  __syncthreads();         // ... and everyone else's
  for (int i = tid; i < SN; i += NTHREADS)
    sGray[i] = 0.299f * sR[i] + 0.587f * sG[i] + 0.114f * sB[i];
#endif
  __syncthreads();
}

// 3x3 Sobel magnitude centered at tile-local pixel (ty,tx); sGray has +1 halo.
__device__ __forceinline__ float sobel_mag(const float* sGray, int ty, int tx) {
  const float* r0 = sGray + ty * SW + tx;
  const float* r1 = r0 + SW;
  const float* r2 = r1 + SW;
  float a00 = r0[0], a01 = r0[1], a02 = r0[2];
  float a10 = r1[0],               a12 = r1[2];
  float a20 = r2[0], a21 = r2[1], a22 = r2[2];
  float gx = (a02 - a00) + 2.0f * (a12 - a10) + (a22 - a20);
  float gy = (a20 - a00) + 2.0f * (a21 - a01) + (a22 - a02);
  return sqrtf(gx * gx + gy * gy);
}

// ---------------- kernels ----------------
__global__ __launch_bounds__(NTHREADS)
void k_minmax(const float* __restrict__ im, unsigned* __restrict__ mm) {
  __shared__ float sR[SN], sG[SN], sB[SN], sGray[SN];
  __shared__ float sMin[NTHREADS / 32], sMax[NTHREADS / 32];

  const int tx = threadIdx.x, ty = threadIdx.y;
  const int tid = ty * TX + tx;
  const int x0 = blockIdx.x * TX, y0 = blockIdx.y * TY, b = blockIdx.z;

  stage_tile(im, b, x0, y0, tid, sR, sG, sB, sGray);

  const int x = x0 + tx, y = y0 + ty;
  float lmin = __builtin_inff();
  float lmax = 0.0f;
  if (x >= 1 && x <= Wn - 2 && y >= 1 && y <= Hn - 2) {
    float m = sobel_mag(sGray, ty, tx);
    lmin = m; lmax = m;
  }
  // wave32 butterfly reduction
  #pragma unroll
  for (int o = 16; o > 0; o >>= 1) {
    lmin = fminf(lmin, __shfl_xor(lmin, o, 32));
    lmax = fmaxf(lmax, __shfl_xor(lmax, o, 32));
  }
  if ((tid & 31) == 0) { sMin[tid >> 5] = lmin; sMax[tid >> 5] = lmax; }
  __syncthreads();
  if (tid == 0) {
    float m0 = sMin[0], m1 = sMax[0];
    #pragma unroll
    for (int i = 1; i < NTHREADS / 32; ++i) {
      m0 = fminf(m0, sMin[i]);
      m1 = fmaxf(m1, sMax[i]);
    }
    atomicMin(&mm[0], __float_as_uint(m0));   // bits monotone: m >= 0
    atomicMax(&mm[1], __float_as_uint(m1));
  }
}

__global__ __launch_bounds__(NTHREADS)
void k_write(const float* __restrict__ im, float* __restrict__ out,
             const unsigned* __restrict__ mm) {
  __shared__ float sR[SN], sG[SN], sB[SN], sGray[SN];

  const int tx = threadIdx.x, ty = threadIdx.y;
  const int tid = ty * TX + tx;
  const int x0 = blockIdx.x * TX, y0 = blockIdx.y * TY, b = blockIdx.z;

  stage_tile(im, b, x0, y0, tid, sR, sG, sB, sGray);

  const float mn = __uint_as_float(mm[0]);
  const float mx = __uint_as_float(mm[1]);
  const float sc = 1.0f / (mx - mn);

  const int x = x0 + tx, y = y0 + ty;
  const int si = (ty + 1) * SW + (tx + 1);     // own pixel (never clamped)
  const float r  = sR[si];
  const float g  = sG[si];
  const float bl = sB[si];

  const size_t ob = (size_t)b * OCHWn + (size_t)y * Wn + x;
  // 128 MiB streamed output: non-temporal so im stays resident in L2
  __builtin_nontemporal_store(r,  &out[ob]);
  __builtin_nontemporal_store(g,  &out[ob + HWn]);
  __builtin_nontemporal_store(bl, &out[ob + 2 * HWn]);

  float nv = 0.0f;
  if (x >= 1 && x <= Wn - 2 && y >= 1 && y <= Hn - 2) {
    float m = sobel_mag(sGray, ty, tx);
    nv = (m - mn) * sc;
  }
  __builtin_nontemporal_store(nv, &out[ob + 3 * HWn]);
}

__global__ void k_init(unsigned* mm) {
  mm[0] = 0x7F800000u;  // +inf  (running min, as monotone uint bits)
  mm[1] = 0x00000000u;  // 0     (running max; mag >= 0)
}

// ---------------- launcher ----------------
extern "C" void kernel_launch(void* const* d_in, const int* in_sizes, int n_in,
                              void* d_out, int out_size, void* d_ws, size_t ws_size,
                              hipStream_t stream) {
  (void)in_sizes; (void)n_in; (void)out_size; (void)ws_size;
  const float* im = (const float*)d_in[0];
  float* out = (float*)d_out;
  unsigned* mm = (unsigned*)d_ws;      // [0]=min bits, [1]=max bits

  dim3 block(TX, TY, 1);
  dim3 grid(Wn / TX, Hn / TY, Bn);     // 16 x 64 x 32

  k_init<<<1, 1, 0, stream>>>(mm);
  k_minmax<<<grid, block, 0, stream>>>(im, mm);
  k_write<<<grid, block, 0, stream>>>(im, out, mm);
}